// Wavelet_60438779789357
// MI455X (gfx1250) — compile-verified
//
#include <hip/hip_runtime.h>

// db2 horizontal DWT, depthwise, stride 2, reflect pad (2,1) on W.
// x: (16,64,256,256) f32 ; w_low/w_high: (64,1,1,4) f32
// out: low (16,64,256,128) then high (16,64,256,128), concatenated.

typedef __attribute__((ext_vector_type(2))) float v2f;
typedef __attribute__((ext_vector_type(8))) float v8f;

#define ROWS_TOTAL (16 * 64 * 256)  // 262144 image rows
#define W_IN 256
#define W_OUT 128
#define WAVES_PER_BLOCK 8
// per-wave LDS slice layout (floats):
//   [0..1]   unused (alignment)
//   [2..3]   xp[0], xp[1]  (reflected left pad: x[2], x[1])
//   [4..259] x[0..255]     (so xp[j] lives at slice[j+2] for j>=2 as well)
// 264 floats = 1056 B per slice -> 16B aligned slices, async b128 dest 16B aligned.
#define SMEM_ROW 264

__global__ __launch_bounds__(256) void dwt_db2_wmma_kernel(
    const float* __restrict__ x,
    const float* __restrict__ wlo,   // (64,1,1,4)
    const float* __restrict__ whi,   // (64,1,1,4)
    float* __restrict__ out)
{
    __shared__ float smem[WAVES_PER_BLOCK][SMEM_ROW];

    const int lane = threadIdx.x & 31;
    const int wave = threadIdx.x >> 5;
    const int row  = blockIdx.x * WAVES_PER_BLOCK + wave;   // (n*64 + c)*256 + h
    const int chan = (row >> 8) & 63;

    const float* xrow  = x + (size_t)row * W_IN;
    float*       smrow = &smem[wave][0];

    // ---- async copy x[0..255] -> smem slice [4..259] (2 x b128 per wave) ----
    unsigned lds_addr = (unsigned)(uintptr_t)(smrow + 4) + (unsigned)lane * 16u;
    unsigned long long gaddr = (unsigned long long)(uintptr_t)(xrow + lane * 4);
    asm volatile("global_load_async_to_lds_b128 %0, %1, off"
                 :: "v"(lds_addr), "v"(gaddr) : "memory");
    lds_addr += 32u * 16u;
    gaddr    += 32ull * 16ull;
    asm volatile("global_load_async_to_lds_b128 %0, %1, off"
                 :: "v"(lds_addr), "v"(gaddr) : "memory");

    // ---- reflect left pad: xp[0]=x[2], xp[1]=x[1] (disjoint from async dest) ----
    if (lane < 2) smrow[2 + lane] = xrow[2 - lane];

    // ---- A matrix (16x4 f32): row M=0 = w_low, row M=8 = w_high ----
    // layout: lanes 0-15 -> K=0 (a.x), K=1 (a.y); lanes 16-31 -> K=2 (a.x), K=3 (a.y)
    const float* wl = wlo + chan * 4;
    const float* wh = whi + chan * 4;
    v2f a = {0.0f, 0.0f};
    if      (lane == 0)  { a.x = wl[0]; a.y = wl[1]; }
    else if (lane == 16) { a.x = wl[2]; a.y = wl[3]; }
    else if (lane == 8)  { a.x = wh[0]; a.y = wh[1]; }
    else if (lane == 24) { a.x = wh[2]; a.y = wh[3]; }

    // ---- wait for async staging before LDS reads ----
    asm volatile("s_wait_asynccnt 0" ::: "memory");

    // Output: D VGPR0 = lanes 0-15: M=0 (low, N=lane); lanes 16-31: M=8 (high, N=lane-16)
    float* olow  = out;
    float* ohigh = out + (size_t)ROWS_TOTAL * W_OUT;
    const int n = lane & 15;
    float* obase = ((lane < 16) ? olow : ohigh) + (size_t)row * W_OUT + n;

    #pragma unroll
    for (int base = 0; base < W_OUT; base += 16) {
        // B (4x16 f32) im2col: lane n: (xp[2(base+n)], xp[2(base+n)+1]) -> K=0,1
        //                      lane 16+n: pair shifted by one position     -> K=2,3
        const int idx = base + n + ((lane < 16) ? 0 : 1);
        v2f b;
        b.x = smrow[2 * idx + 2];   // xp[2*idx]
        b.y = smrow[2 * idx + 3];   // xp[2*idx+1]

        v8f c = {};
        c = __builtin_amdgcn_wmma_f32_16x16x4_f32(
                /*neg_a=*/false, a, /*neg_b=*/false, b,
                /*c_mod=*/(short)0, c, /*reuse_a=*/false, /*reuse_b=*/false);

        obase[base] = c[0];   // low for lanes 0-15, high for lanes 16-31
    }
}

extern "C" void kernel_launch(void* const* d_in, const int* in_sizes, int n_in,
                              void* d_out, int out_size, void* d_ws, size_t ws_size,
                              hipStream_t stream) {
    const float* x   = (const float*)d_in[0];
    const float* wlo = (const float*)d_in[1];
    const float* whi = (const float*)d_in[2];
    float* out = (float*)d_out;
    (void)in_sizes; (void)n_in; (void)out_size; (void)d_ws; (void)ws_size;

    dim3 grid(ROWS_TOTAL / WAVES_PER_BLOCK);  // 32768 blocks
    dim3 block(256);                          // 8 waves per block
    dwt_db2_wmma_kernel<<<grid, block, 0, stream>>>(x, wlo, whi, out);
}